// GCN_25091198943652
// MI455X (gfx1250) — compile-verified
//
#include <hip/hip_runtime.h>

#define N_NODES 50000
#define N_EDGES 800000
#define IN_C    64
#define HID_C   128
#define OUT_C   64

typedef __attribute__((ext_vector_type(2))) float v2f;
typedef __attribute__((ext_vector_type(8))) float v8f;

// ---------------- degree / normalization ----------------

__global__ void gcn_deg_init(float* __restrict__ deg) {
    int n = blockIdx.x * blockDim.x + threadIdx.x;
    if (n < N_NODES) deg[n] = 1.0f;   // self-loop contributes 1
}

__global__ void gcn_deg_accum(const int* __restrict__ dst, float* __restrict__ deg) {
    int e = blockIdx.x * blockDim.x + threadIdx.x;
    if (e < N_EDGES) atomicAdd(&deg[dst[e]], 1.0f);
}

__global__ void gcn_deg_to_dinv(float* __restrict__ deg) {
    int n = blockIdx.x * blockDim.x + threadIdx.x;
    if (n < N_NODES) deg[n] = rsqrtf(deg[n]);  // deg >= 1 always (self loops)
}

// ---------------- GEMM1: xw = x(50000x64) @ W1(64x128), fp32 WMMA ----------------
// block = 256 threads = 8 waves; block handles a 16x128 row-stripe of the output.
// wave w computes the 16x16 tile at columns [16w, 16w+16).

__global__ __launch_bounds__(256) void gcn_gemm1_wmma(
    const float* __restrict__ x, const float* __restrict__ W1, float* __restrict__ xw) {
    const int lane = threadIdx.x & 31;
    const int wave = threadIdx.x >> 5;
    const int m0   = blockIdx.x * 16;     // 50000 = 3125 * 16, no tail
    const int n0   = wave * 16;
    const int half = lane >> 4;           // 0: lanes 0-15, 1: lanes 16-31
    const int l15  = lane & 15;
    const int row  = m0 + l15;            // A: M = lane (both halves)
    const int col  = n0 + l15;            // B/C/D: N = lane
    v8f acc = {0.f, 0.f, 0.f, 0.f, 0.f, 0.f, 0.f, 0.f};

    for (int k0 = 0; k0 < IN_C; k0 += 4) {
        const int ka = k0 + half * 2;     // A/B: halves hold K pairs {k0,k0+1} / {k0+2,k0+3}
        v2f a = *(const v2f*)(x + row * IN_C + ka);          // A[row][ka], A[row][ka+1]
        v2f b;
        b.x = W1[ka * HID_C + col];                          // B[ka][col]
        b.y = W1[(ka + 1) * HID_C + col];                    // B[ka+1][col]
        acc = __builtin_amdgcn_wmma_f32_16x16x4_f32(
            /*neg_a=*/false, a, /*neg_b=*/false, b,
            /*c_mod=*/(short)0, acc, /*reuse_a=*/false, /*reuse_b=*/false);
    }
#pragma unroll
    for (int r = 0; r < 8; ++r)           // D: VGPR r holds M = half*8 + r
        xw[(size_t)(m0 + half * 8 + r) * HID_C + col] = acc[r];
}

// ---------------- GEMM2: xw2 = relu(h)(50000x128) @ W2(128x64), ReLU fused into A ----------------
// block = 128 threads = 4 waves (64 output columns).

__global__ __launch_bounds__(128) void gcn_gemm2_wmma(
    const float* __restrict__ h, const float* __restrict__ W2, float* __restrict__ xw) {
    const int lane = threadIdx.x & 31;
    const int wave = threadIdx.x >> 5;
    const int m0   = blockIdx.x * 16;
    const int n0   = wave * 16;
    const int half = lane >> 4;
    const int l15  = lane & 15;
    const int row  = m0 + l15;
    const int col  = n0 + l15;
    v8f acc = {0.f, 0.f, 0.f, 0.f, 0.f, 0.f, 0.f, 0.f};

    for (int k0 = 0; k0 < HID_C; k0 += 4) {
        const int ka = k0 + half * 2;
        v2f a = *(const v2f*)(h + (size_t)row * HID_C + ka);
        a.x = fmaxf(a.x, 0.0f);           // fused ReLU
        a.y = fmaxf(a.y, 0.0f);
        v2f b;
        b.x = W2[ka * OUT_C + col];
        b.y = W2[(ka + 1) * OUT_C + col];
        acc = __builtin_amdgcn_wmma_f32_16x16x4_f32(
            false, a, false, b, (short)0, acc, false, false);
    }
#pragma unroll
    for (int r = 0; r < 8; ++r)
        xw[(size_t)(m0 + half * 8 + r) * OUT_C + col] = acc[r];
}

// ---------------- aggregation init: agg[n][c] = b[c] + xw[n][c]*dinv[n]^2 (self loop) ----------------

template <int C>
__global__ void gcn_agg_init(const float* __restrict__ xw, const float* __restrict__ dinv,
                             const float* __restrict__ bias, float* __restrict__ agg) {
    int i = blockIdx.x * blockDim.x + threadIdx.x;
    if (i >= N_NODES * C) return;
    int n = i / C;
    int c = i % C;
    float dn = dinv[n];
    agg[i] = bias[c] + xw[i] * dn * dn;
}

// ---------------- edge scatter: agg[dst] += xw[src] * dinv[src]*dinv[dst] ----------------
// One edge per C/4 lanes; each lane gathers a float4 (coalesced 16B) and does 4 f32 atomics.

template <int C>
__global__ void gcn_scatter(const int* __restrict__ src, const int* __restrict__ dst,
                            const float* __restrict__ dinv,
                            const float* __restrict__ xw, float* __restrict__ agg) {
    const int LPE = C / 4;                        // lanes per edge (32 or 16)
    int t = blockIdx.x * blockDim.x + threadIdx.x;
    int e = t / LPE;
    if (e >= N_EDGES) return;
    int q = t % LPE;
    int s = src[e];
    int d = dst[e];
    float norm = dinv[s] * dinv[d];
    const float4 v = *(const float4*)(xw + (size_t)s * C + q * 4);
    float* base = agg + (size_t)d * C + q * 4;
    atomicAdd(base + 0, v.x * norm);
    atomicAdd(base + 1, v.y * norm);
    atomicAdd(base + 2, v.z * norm);
    atomicAdd(base + 3, v.w * norm);
}

// ---------------- launch ----------------

extern "C" void kernel_launch(void* const* d_in, const int* in_sizes, int n_in,
                              void* d_out, int out_size, void* d_ws, size_t ws_size,
                              hipStream_t stream) {
    const float* x  = (const float*)d_in[0];            // [50000,64]
    const int*   ei = (const int*)d_in[1];              // [2,800000] (JAX default int32)
    const float* W1 = (const float*)d_in[2];            // [64,128]
    const float* b1 = (const float*)d_in[3];            // [128]
    const float* W2 = (const float*)d_in[4];            // [128,64]
    const float* b2 = (const float*)d_in[5];            // [64]
    float* out = (float*)d_out;                         // [50000,64]

    const int* src = ei;                                // edge_index[0]
    const int* dst = ei + N_EDGES;                      // edge_index[1]

    // workspace carve-up (floats): dinv | xw1 (reused as xw2) | agg1
    float* dinv = (float*)d_ws;                         // 50176 (padded, keeps 256B alignment)
    float* xw1  = dinv + 50176;                         // 50000*128 = 6,400,000
    float* agg1 = xw1 + 6400000;                        // 50000*128 = 6,400,000
    float* xw2  = xw1;                                  // reuse: xw1 dead after scatter1

    // 1) symmetric normalization  dinv = rsqrt(deg(A+I))
    gcn_deg_init<<<(N_NODES + 255) / 256, 256, 0, stream>>>(dinv);
    gcn_deg_accum<<<(N_EDGES + 255) / 256, 256, 0, stream>>>(dst, dinv);
    gcn_deg_to_dinv<<<(N_NODES + 255) / 256, 256, 0, stream>>>(dinv);

    // 2) layer 1: xw1 = x @ W1   (WMMA f32 16x16x4)
    gcn_gemm1_wmma<<<N_NODES / 16, 256, 0, stream>>>(x, W1, xw1);

    // 3) aggregate layer 1 (self-loop + bias folded into init)
    gcn_agg_init<HID_C><<<(N_NODES * HID_C + 255) / 256, 256, 0, stream>>>(xw1, dinv, b1, agg1);
    gcn_scatter<HID_C><<<(N_EDGES * (HID_C / 4) + 255) / 256, 256, 0, stream>>>(src, dst, dinv, xw1, agg1);

    // 4) layer 2: xw2 = relu(agg1) @ W2  (ReLU fused into WMMA A-load)
    gcn_gemm2_wmma<<<N_NODES / 16, 128, 0, stream>>>(agg1, W2, xw2);

    // 5) aggregate layer 2 directly into d_out
    gcn_agg_init<OUT_C><<<(N_NODES * OUT_C + 255) / 256, 256, 0, stream>>>(xw2, dinv, b2, out);
    gcn_scatter<OUT_C><<<(N_EDGES * (OUT_C / 4) + 255) / 256, 256, 0, stream>>>(src, dst, dinv, xw2, out);
}